// LennardJonesVerlet_5695126634505
// MI455X (gfx1250) — compile-verified
//
#include <hip/hip_runtime.h>

typedef __attribute__((ext_vector_type(2))) float v2f;
typedef __attribute__((ext_vector_type(8))) float v8f;
typedef __attribute__((ext_vector_type(4))) int   v4i;

#define LJ_THREADS 256
#define LJ_BLOCKS  2048

struct F3 { float x, y, z; };

// One LJ pair energy term (without the 4*eps prefactor, applied at the end).
__device__ __forceinline__ float lj_pair(const F3* __restrict__ xv, int i, int j,
                                         float Lx, float Ly, float Lz,
                                         float iLx, float iLy, float iLz,
                                         float sigma2) {
    F3 a = xv[i];
    F3 b = xv[j];
    float dx = a.x - b.x;
    float dy = a.y - b.y;
    float dz = a.z - b.z;
    // minimum image: diff -= L * round(diff / L); rintf == round-half-even (v_rndne_f32)
    dx -= Lx * rintf(dx * iLx);
    dy -= Ly * rintf(dy * iLy);
    dz -= Lz * rintf(dz * iLz);
    float r2 = dx * dx + dy * dy + dz * dz;
    float s2 = sigma2 / r2;          // IEEE division for fp32 fidelity
    float s6 = s2 * s2 * s2;
    return s6 * s6 - s6;
}

__global__ __launch_bounds__(LJ_THREADS)
void lj_pairs_kernel(const float* __restrict__ x,
                     const float* __restrict__ sigma_p,
                     const float* __restrict__ box,
                     const int* __restrict__ idx_i,
                     const int* __restrict__ idx_j,
                     float* __restrict__ partials,
                     int n_pairs) {
    const float Lx = box[0], Ly = box[1], Lz = box[2];
    const float iLx = 1.0f / Lx, iLy = 1.0f / Ly, iLz = 1.0f / Lz;
    const float sg = sigma_p[0];
    const float sigma2 = sg * sg;

    const F3* xv = reinterpret_cast<const F3*>(x);
    const v4i* pi = reinterpret_cast<const v4i*>(idx_i);
    const v4i* pj = reinterpret_cast<const v4i*>(idx_j);

    const int tid = blockIdx.x * blockDim.x + threadIdx.x;
    const int T = gridDim.x * blockDim.x;
    const int chunks = n_pairs >> 2;

    float acc = 0.0f;
    for (int c = tid; c < chunks; c += T) {
        // streaming index reads: non-temporal so they don't evict the hot 3MB x array
        v4i vi = __builtin_nontemporal_load(pi + c);
        v4i vj = __builtin_nontemporal_load(pj + c);
#pragma unroll
        for (int u = 0; u < 4; ++u) {
            acc += lj_pair(xv, vi[u], vj[u], Lx, Ly, Lz, iLx, iLy, iLz, sigma2);
        }
    }
    // generic tail (empty for 8,388,608 pairs)
    if (tid == 0) {
        for (int p = chunks << 2; p < n_pairs; ++p) {
            acc += lj_pair(xv, idx_i[p], idx_j[p], Lx, Ly, Lz, iLx, iLy, iLz, sigma2);
        }
    }

    // deterministic in-block reduction: wave32 shuffle tree, then LDS across 8 waves
    float v = acc;
#pragma unroll
    for (int off = 16; off > 0; off >>= 1) v += __shfl_down(v, off, 32);

    __shared__ float smem[LJ_THREADS / 32];
    const int lane = threadIdx.x & 31;
    const int wid = threadIdx.x >> 5;
    if (lane == 0) smem[wid] = v;
    __syncthreads();
    if (threadIdx.x == 0) {
        float b = 0.0f;
#pragma unroll
        for (int w = 0; w < LJ_THREADS / 32; ++w) b += smem[w];
        partials[blockIdx.x] = b;
    }
}

// Final reduction on a single wave of 32 lanes. The 32 lane partials are
// collapsed with V_WMMA_F32_16X16X4_F32: A[m][0]=p_m (lanes 0-15, VGPR0),
// A[m][2]=p_{m+16} (lanes 16-31, VGPR0), A[*][1]=A[*][3]=0; B=ones, C=0.
// D[m][n] = p_m + p_{m+16}. Each lane sums its 8 D VGPRs (rows 0-7 on lanes
// 0-15, rows 8-15 on lanes 16-31); total = halfsum(lane0) + halfsum(lane16).
__global__ __launch_bounds__(32)
void lj_reduce_kernel(const float* __restrict__ partials, int n,
                      const float* __restrict__ eps_p,
                      float* __restrict__ out) {
    const int lane = threadIdx.x;  // 0..31, one full wave, EXEC all ones
    float p = 0.0f;
    for (int k = lane; k < n; k += 32) p += partials[k];  // uniform trip count

    v2f a; a[0] = p;    a[1] = 0.0f;
    v2f b; b[0] = 1.0f; b[1] = 1.0f;
    v8f c = {};
    v8f d = __builtin_amdgcn_wmma_f32_16x16x4_f32(
        /*neg_a=*/false, a, /*neg_b=*/false, b,
        /*c_mod=*/(short)0, c, /*reuse_a=*/false, /*reuse_b=*/false);

    float s = d[0] + d[1] + d[2] + d[3] + d[4] + d[5] + d[6] + d[7];
    float tot = __shfl(s, 0, 32) + __shfl(s, 16, 32);
    if (lane == 0) out[0] = 4.0f * eps_p[0] * tot;
}

extern "C" void kernel_launch(void* const* d_in, const int* in_sizes, int n_in,
                              void* d_out, int out_size, void* d_ws, size_t ws_size,
                              hipStream_t stream) {
    // setup_inputs order: x, eps, sigma, box_size, idx_i, idx_j
    const float* x       = (const float*)d_in[0];
    const float* eps_p   = (const float*)d_in[1];
    const float* sigma_p = (const float*)d_in[2];
    const float* box     = (const float*)d_in[3];
    const int*   idx_i   = (const int*)d_in[4];
    const int*   idx_j   = (const int*)d_in[5];
    const int n_pairs = in_sizes[4];

    float* partials = (float*)d_ws;  // LJ_BLOCKS floats = 8 KB scratch

    lj_pairs_kernel<<<LJ_BLOCKS, LJ_THREADS, 0, stream>>>(
        x, sigma_p, box, idx_i, idx_j, partials, n_pairs);
    lj_reduce_kernel<<<1, 32, 0, stream>>>(partials, LJ_BLOCKS, eps_p, (float*)d_out);
}